// BispectrumS1xS1_45810121179817
// MI455X (gfx1250) — compile-verified
//
#include <hip/hip_runtime.h>
#include <hip/hip_bf16.h>

typedef float v2f __attribute__((ext_vector_type(2)));
typedef float v8f __attribute__((ext_vector_type(8)));

// ---------------------------------------------------------------------------
// Stage 1: 64x64 2D FFT per batch as DFT matmuls on the WMMA f32 16x16x4 path.
//   Y = F * x   (F = C + iS,  C[m,k]=cos(2*pi*m*k/64), S = -sin(...))
//   X = Y * F   (F symmetric)
// One block per batch, 16 waves (512 threads); wave w owns output tile
// (w/4, w%4). Twiddles e^{-2*pi*i*k/64} precomputed in a 64-entry LDS table,
// coefficients indexed by (m*k) mod 64. Intermediate Y kept in LDS.
// Output X written interleaved (re,im) to workspace.
// ---------------------------------------------------------------------------
__global__ __launch_bounds__(512) void fft2_wmma_kernel(
    const float* __restrict__ x, float* __restrict__ Xout) {
  __shared__ float xs[64 * 64];
  __shared__ float yr[64 * 64];
  __shared__ float yi[64 * 64];
  __shared__ float2 tw[64];   // tw[k] = (cos(-2pi k/64), sin(-2pi k/64))

  const int batch = blockIdx.x;
  const int tid = threadIdx.x;

  // Load x (16 KB) into LDS with 16B vectors; build twiddle table.
  {
    const float4* xg = (const float4*)(x + batch * 4096);
    float4* xl = (float4*)xs;
    for (int i = tid; i < 1024; i += 512) xl[i] = xg[i];
    if (tid < 64) {
      const float ang = (float)tid * (-6.2831853071795864769f / 64.0f);
      tw[tid] = make_float2(cosf(ang), sinf(ang));
    }
  }
  __syncthreads();

  const int lane = tid & 31;
  const int wave = tid >> 5;
  const int half = lane >> 4;   // which half-wave
  const int nl   = lane & 15;   // N within tile (B/C/D operands)
  const int mrow = lane & 15;   // M within tile (A operand)
  const int m0 = (wave >> 2) << 4;
  const int n0 = (wave & 3) << 4;

  // ---- Stage A: Yr = C*x, Yi = S*x ----
  v8f accr = {};
  v8f acci = {};
  for (int k0 = 0; k0 < 64; k0 += 4) {
    const int m  = m0 + mrow;
    const int ka = k0 + 2 * half;        // A: VGPR0 holds K=ka, VGPR1 K=ka+1
    const float2 t0 = tw[(m * ka) & 63];
    const float2 t1 = tw[(m * (ka + 1)) & 63];
    v2f ac, as_;
    ac.x = t0.x;  ac.y = t1.x;           // cos(-2pi mk/64)
    as_.x = t0.y; as_.y = t1.y;          // sin(-2pi mk/64)

    v2f bx;                              // B = x tile (4x16)
    bx.x = xs[(ka)     * 64 + n0 + nl];
    bx.y = xs[(ka + 1) * 64 + n0 + nl];
    accr = __builtin_amdgcn_wmma_f32_16x16x4_f32(false, ac,  false, bx,
                                                 (short)0, accr, false, false);
    acci = __builtin_amdgcn_wmma_f32_16x16x4_f32(false, as_, false, bx,
                                                 (short)0, acci, false, false);
  }
  // Spill Y tiles to LDS for the second (right-side) matmul.
  for (int r = 0; r < 8; ++r) {
    const int row = m0 + r + 8 * half;
    const int col = n0 + nl;
    yr[row * 64 + col] = accr[r];
    yi[row * 64 + col] = acci[r];
  }
  __syncthreads();

  // ---- Stage B: Xr = Yr*C - Yi*S,  Xi = Yr*S + Yi*C ----
  v8f xr = {};
  v8f xi = {};
  for (int k0 = 0; k0 < 64; k0 += 4) {
    const int ka = k0 + 2 * half;
    v2f ayr, ayi;                        // A = Y tile (16x4) from LDS
    ayr.x = yr[(m0 + mrow) * 64 + ka];
    ayr.y = yr[(m0 + mrow) * 64 + ka + 1];
    ayi.x = yi[(m0 + mrow) * 64 + ka];
    ayi.y = yi[(m0 + mrow) * 64 + ka + 1];

    const int n = n0 + nl;               // B = F coefficient tile (4x16)
    const float2 t0 = tw[(ka * n) & 63];
    const float2 t1 = tw[((ka + 1) * n) & 63];
    v2f bc, bs, bns;
    bc.x = t0.x;   bc.y = t1.x;
    bs.x = t0.y;   bs.y = t1.y;
    bns.x = -t0.y; bns.y = -t1.y;

    xr = __builtin_amdgcn_wmma_f32_16x16x4_f32(false, ayr, false, bc,
                                               (short)0, xr, false, false);
    xr = __builtin_amdgcn_wmma_f32_16x16x4_f32(false, ayi, false, bns,
                                               (short)0, xr, false, false);
    xi = __builtin_amdgcn_wmma_f32_16x16x4_f32(false, ayr, false, bs,
                                               (short)0, xi, false, false);
    xi = __builtin_amdgcn_wmma_f32_16x16x4_f32(false, ayi, false, bc,
                                               (short)0, xi, false, false);
  }

  // Write X interleaved (re,im) to workspace.
  float2* Xo = (float2*)Xout + batch * 4096;
  for (int r = 0; r < 8; ++r) {
    const int row = m0 + r + 8 * half;
    const int col = n0 + nl;
    Xo[row * 64 + col] = make_float2(xr[r], xi[r]);
  }
}

// ---------------------------------------------------------------------------
// Stage 2: B[p,q] = X[p] * X[q] * conj(X[p+q])   (indices mod 64 per axis)
// One block per (batch, p): stage the whole 32 KB X in LDS, stream the 32 KB
// output row with coalesced float4 (b128) stores. Output-write bound:
// 256 MB total -> ~11 us floor at 23.3 TB/s HBM.
// ---------------------------------------------------------------------------
__global__ __launch_bounds__(256) void bispectrum_kernel(
    const float* __restrict__ Xw, float* __restrict__ out) {
  __shared__ float2 Xs[4096];

  const int p = blockIdx.x;
  const int batch = blockIdx.y;
  const int tid = threadIdx.x;

  {
    const float4* src = (const float4*)(Xw + (size_t)batch * 8192);
    float4* dst = (float4*)Xs;
    for (int i = tid; i < 2048; i += 256) dst[i] = src[i];
  }
  __syncthreads();

  const int a = p >> 6;
  const int b = p & 63;
  const float2 Xp = Xs[p];

  float4* outv = (float4*)(out + ((size_t)(batch * 4096 + p) * 4096) * 2);

  for (int it = 0; it < 8; ++it) {
    const int q0 = it * 512 + tid * 2;
    float4 res;
    {
      const int q = q0;
      const int c = q >> 6, d = q & 63;
      const float2 Xq = Xs[q];
      const float2 Xt = Xs[(((a + c) & 63) << 6) | ((b + d) & 63)];
      const float mr = Xp.x * Xq.x - Xp.y * Xq.y;
      const float mi = Xp.x * Xq.y + Xp.y * Xq.x;
      res.x = mr * Xt.x + mi * Xt.y;   // * conj(Xt)
      res.y = mi * Xt.x - mr * Xt.y;
    }
    {
      const int q = q0 + 1;
      const int c = q >> 6, d = q & 63;
      const float2 Xq = Xs[q];
      const float2 Xt = Xs[(((a + c) & 63) << 6) | ((b + d) & 63)];
      const float mr = Xp.x * Xq.x - Xp.y * Xq.y;
      const float mi = Xp.x * Xq.y + Xp.y * Xq.x;
      res.z = mr * Xt.x + mi * Xt.y;
      res.w = mi * Xt.x - mr * Xt.y;
    }
    outv[q0 >> 1] = res;               // 16B coalesced store
  }
}

extern "C" void kernel_launch(void* const* d_in, const int* in_sizes, int n_in,
                              void* d_out, int out_size, void* d_ws, size_t ws_size,
                              hipStream_t stream) {
  (void)in_sizes; (void)n_in; (void)out_size; (void)ws_size;
  const float* x = (const float*)d_in[0];  // (2, 64, 64) f32
  float* out = (float*)d_out;              // (2, 4096, 4096) complex64 interleaved
  float* Xw = (float*)d_ws;                // 64 KB: X per batch, interleaved

  fft2_wmma_kernel<<<2, 512, 0, stream>>>(x, Xw);
  bispectrum_kernel<<<dim3(4096, 2), 256, 0, stream>>>(Xw, out);
}